// InvertedReorg_59596966199519
// MI455X (gfx1250) — compile-verified
//
#include <hip/hip_runtime.h>
#include <stdint.h>

// InvertedReorg (depth-to-space, s=2):
//   in  : (32, 256, 64, 64) f32
//   out : (32,  64, 128,128) f32
//   out[b][c][s1*64+hh][s2*64+ww] = in[b][(2*s1+s2)*64+c][hh][ww]
//
// Pure permutation: 0 FLOPs, 256 MiB traffic -> ~11us at 23.3 TB/s (HBM bound).
// Use the CDNA5 async LDS data mover: GLOBAL_LOAD_ASYNC_TO_LDS_B128 /
// GLOBAL_STORE_ASYNC_FROM_LDS_B128 (ASYNCcnt), 16B per lane, fully coalesced
// 256B contiguous runs on both sides.

typedef float f4 __attribute__((ext_vector_type(4)));

#define THREADS 256
#define K 4  // float4 tiles per thread

__global__ __launch_bounds__(THREADS, 1)
void InvertedReorg_async_kernel(const f4* __restrict__ in, f4* __restrict__ out) {
    // Lane-private staging slots: smem[k*256 + tid], 16B each (16KB/block).
    __shared__ f4 smem[THREADS * K];

    const unsigned tid = threadIdx.x;
    // Flat shared pointer: addr[31:0] is the LDS byte offset (aperture layout).
    const unsigned lds0 = (unsigned)(uintptr_t)(&smem[tid]);

    // Output float4 index space: o = ((b*64 + c)*128 + H)*32 + W4
    const unsigned o0 = blockIdx.x * (THREADS * K) + tid;

    const f4* gin[K];
    f4*       gout[K];
    unsigned  ldso[K];

#pragma unroll
    for (int k = 0; k < K; ++k) {
        const unsigned o  = o0 + (unsigned)k * THREADS;
        const unsigned W4 = o & 31u;            // 32 float4 per output row (W=128)
        const unsigned H  = (o >> 5) & 127u;    // 128 rows
        const unsigned c  = (o >> 12) & 63u;    // 64 output channels
        const unsigned b  = o >> 18;            // 32 batches
        const unsigned s2 = W4 >> 4;            // quadrant (W)
        const unsigned w4 = W4 & 15u;           // 16 float4 per 64-float run
        const unsigned s1 = H >> 6;             // quadrant (H)
        const unsigned hh = H & 63u;
        const unsigned ch = (((s1 << 1) | s2) << 6) | c;      // (2*s1+s2)*64 + c
        const unsigned i4 = (((b << 8) | ch) << 10) | (hh << 4) | w4; // ((b*256+ch)*64+hh)*16+w4

        gin[k]  = in + i4;
        gout[k] = out + o;
        ldso[k] = lds0 + (unsigned)(k * THREADS * sizeof(f4));
    }

    // Stage 1: async global -> LDS, 16B per lane per op (ASYNCcnt += K)
#pragma unroll
    for (int k = 0; k < K; ++k) {
        asm volatile("global_load_async_to_lds_b128 %0, %1, off"
                     :: "v"(ldso[k]), "v"(gin[k]) : "memory");
    }
    // Wait for all async loads to land in LDS.
    asm volatile("s_wait_asynccnt 0x0" ::: "memory");

    // Stage 2: async LDS -> global, per-lane scatter addresses.
#pragma unroll
    for (int k = 0; k < K; ++k) {
        asm volatile("global_store_async_from_lds_b128 %0, %1, off"
                     :: "v"(gout[k]), "v"(ldso[k]) : "memory");
    }
    // Drain before wave end (S_ENDPGM also does an implicit wait-idle).
    asm volatile("s_wait_asynccnt 0x0" ::: "memory");
}

extern "C" void kernel_launch(void* const* d_in, const int* in_sizes, int n_in,
                              void* d_out, int out_size, void* d_ws, size_t ws_size,
                              hipStream_t stream) {
    (void)in_sizes; (void)n_in; (void)d_ws; (void)ws_size;
    const f4* in  = (const f4*)d_in[0];
    f4*       out = (f4*)d_out;

    // out_size = 32*64*128*128 = 2^25 floats = 2^23 float4s.
    const int total4 = out_size / 4;
    const int blocks = total4 / (THREADS * K);  // 8192 for the reference shape

    InvertedReorg_async_kernel<<<blocks, THREADS, 0, stream>>>(in, out);
}